// CTCLossLayer_37160057045692
// MI455X (gfx1250) — compile-verified
//
#include <hip/hip_runtime.h>
#include <math.h>

// ---------------------------------------------------------------------------
// CTC loss forward for MI455X (gfx1250, wave32).
//   Kernel 1: per-(b,t) log-sum-exp normalizer. Row sums of exp() via
//             V_WMMA_F32_16X16X4_F32 (A = exp tile, B = ones): exact f32
//             reduction on the matrix pipe. LDS staged with async-to-LDS DMA.
//   Kernel 2: per-batch sequential alpha recursion (S=2L+1 states in LDS,
//             one barrier per time step, async double-buffered pred row).
// ---------------------------------------------------------------------------

#define NEGV (-1.0e30f)

typedef __attribute__((ext_vector_type(2))) float v2f;
typedef __attribute__((ext_vector_type(8))) float v8f;
typedef __attribute__((ext_vector_type(4))) int v4i;

#if __has_builtin(__builtin_amdgcn_global_load_async_to_lds_b128)
#define HAVE_ASYNC_LDS 1
#else
#define HAVE_ASYNC_LDS 0
#endif

#define GLOBAL_AS __attribute__((address_space(1)))
#define LDS_AS __attribute__((address_space(3)))

// Copy 16 bytes / lane from global to LDS via the async-to-LDS engine
// (tracked on ASYNCcnt, no VGPR data round-trip).
__device__ __forceinline__ void async_copy16(const float* g, float* l) {
#if HAVE_ASYNC_LDS
    __builtin_amdgcn_global_load_async_to_lds_b128(
        (GLOBAL_AS v4i*)g, (LDS_AS v4i*)l, 0, 0);
#else
    (void)g; (void)l;
#endif
}

__device__ __forceinline__ void wait_async0() {
#if HAVE_ASYNC_LDS
#if __has_builtin(__builtin_amdgcn_s_wait_asynccnt)
    __builtin_amdgcn_s_wait_asynccnt(0);
#else
    asm volatile("s_wait_asynccnt 0x0" ::: "memory");
#endif
#endif
}

__device__ __forceinline__ float logaddexp3(float a, float b, float c) {
    float m = fmaxf(a, fmaxf(b, c));
    float s = __expf(a - m) + __expf(b - m) + __expf(c - m);
    return m + __logf(s);
}

// ---------------------------------------------------------------------------
// Kernel 1: norm[row] = max_c pred[row,c] + log(sum_c exp(pred[row,c]-max))
// Block = 128 threads (4 waves). Each wave owns 16 rows staged in LDS.
// ---------------------------------------------------------------------------
template <int C>
__global__ __launch_bounds__(128) void lse_norm_kernel(
    const float* __restrict__ pred, float* __restrict__ norm, int nrows) {
    constexpr int RPW = 16;   // rows per wave
    constexpr int WAVES = 4;
    constexpr int PITCH = C + 4;  // LDS pad: rows land on distinct bank groups
    __shared__ float tile[WAVES][RPW][PITCH];

    const int lane = threadIdx.x & 31;
    const int wave = threadIdx.x >> 5;
    const int rowBase = blockIdx.x * (WAVES * RPW) + wave * RPW;
    if (blockIdx.x * (WAVES * RPW) >= nrows) return;  // block-uniform guard

    // ---- stage 16 rows x C floats into LDS ----
#if HAVE_ASYNC_LDS
    for (int i = lane; i < RPW * (C / 4); i += 32) {
        const int rr = i / (C / 4);
        const int cc = (i % (C / 4)) * 4;
        async_copy16(pred + (size_t)(rowBase + rr) * C + cc,
                     &tile[wave][rr][cc]);
    }
    wait_async0();
    __syncthreads();
#else
    for (int i = lane; i < RPW * (C / 4); i += 32) {
        const int rr = i / (C / 4);
        const int cc = (i % (C / 4)) * 4;
        const float4 v =
            *(const float4*)(pred + (size_t)(rowBase + rr) * C + cc);
        *(float4*)&tile[wave][rr][cc] = v;
    }
    __syncthreads();
#endif

    // ---- per-row max: 2 lanes per row (halves), combined via shfl_xor 16 --
    const int r    = lane & 15;
    const int half = lane >> 4;
    float m = -INFINITY;
    const float* rowp = &tile[wave][r][half * (C / 2)];
#pragma unroll
    for (int j = 0; j < C / 2; j += 4) {
        float4 v = *(const float4*)(rowp + j);
        m = fmaxf(m, fmaxf(fmaxf(v.x, v.y), fmaxf(v.z, v.w)));
    }
    m = fmaxf(m, __shfl_xor(m, 16, 32));  // both halves now hold row max

#if __has_builtin(__builtin_amdgcn_wmma_f32_16x16x4_f32)
    // ---- row sums of exp(x - m) via WMMA: D += A(16x4) * ones(4x16) ----
    // A layout (32-bit, 16x4): lanes 0-15 -> M=lane, K={k,k+1};
    //                          lanes 16-31 -> M=lane-16, K={k+2,k+3}.
    v8f acc = {};
    const v2f ones = {1.0f, 1.0f};
    const int cb = half * 2;
#pragma unroll 16
    for (int k = 0; k < C; k += 4) {
        const float* p = &tile[wave][r][k + cb];
        v2f a;
        a.x = __expf(p[0] - m);
        a.y = __expf(p[1] - m);
        acc = __builtin_amdgcn_wmma_f32_16x16x4_f32(
            false, a, false, ones, (short)0, acc, false, false);
    }
    // D layout: lanes 0-15 hold rows v in acc[v], lanes 16-31 hold rows v+8.
    // Every column identical (B = ones) -> any lane's acc[v] is the rowsum.
    // Writers: lanes 0..7 -> rows 0..7, lanes 24..31 -> rows 8..15; each
    // writer's own `m` matches its row.
    int sel = -1;
    if (lane < 8) sel = lane;
    else if (lane >= 24) sel = lane - 24;
    if (sel >= 0) {
        float s = acc[0];
        s = (sel == 1) ? acc[1] : s;
        s = (sel == 2) ? acc[2] : s;
        s = (sel == 3) ? acc[3] : s;
        s = (sel == 4) ? acc[4] : s;
        s = (sel == 5) ? acc[5] : s;
        s = (sel == 6) ? acc[6] : s;
        s = (sel == 7) ? acc[7] : s;
        const int row = rowBase + (lane & 15);
        norm[row] = m + __logf(s);
    }
#else
    // Fallback: plain shuffle reduction.
    float s = 0.0f;
#pragma unroll
    for (int j = 0; j < C / 2; ++j) s += __expf(rowp[j] - m);
    s += __shfl_xor(s, 16, 32);
    if (lane < 16) norm[rowBase + r] = m + __logf(s);
#endif
}

// ---------------------------------------------------------------------------
// Kernel 2: CTC alpha recursion. One workgroup per batch element.
// 544 threads (17 waves) -> one state per thread (S = 513).
// Prediction row (1 KB) double-buffered in LDS; wave 0 drives async DMA.
// ---------------------------------------------------------------------------
__global__ __launch_bounds__(544) void ctc_scan_kernel(
    const float* __restrict__ pred, const float* __restrict__ norm,
    const int* __restrict__ labels, const int* __restrict__ in_len,
    const int* __restrict__ lab_len, float* __restrict__ loss,
    int T, int C, int L) {
    constexpr int SMAX = 513;
    __shared__ float abuf[2][SMAX];
    __shared__ float rowbuf[2][256];
    __shared__ int   extS[SMAX];
    __shared__ unsigned char skipS[SMAX];

    const int b = blockIdx.x;
    const int tid = threadIdx.x;
    const int lane = tid & 31;
    const int S = 2 * L + 1;
    if (S > SMAX) return;  // uniform; never taken for this problem
    const size_t pbase = (size_t)b * T * C;
    const int blank = C - 1;

    // Extended (blank-interleaved) label sequence.
    for (int s = tid; s < S; s += blockDim.x) {
        int e = blank;
        if (s & 1) {
            int v = labels[b * L + (s >> 1)] - 1;  // shift, clamp at 0
            e = v > 0 ? v : 0;
        }
        extS[s] = e;
    }
    __syncthreads();
    for (int s = tid; s < S; s += blockDim.x) {
        // odd states are never blank; skip allowed when label differs from
        // the one two states back (s<3 pads with blank -> allowed).
        skipS[s] = (unsigned char)((s & 1) && (s < 3 || extS[s] != extS[s - 2]));
    }

    int Tin = in_len[b];
    if (Tin > T) Tin = T;
    if (Tin < 1) Tin = 1;
    // Alpha freezes for t >= in_len (block-uniform) -> just stop the loop.

    // ---- stage row 0 (and prefetch row 1) ----
#if HAVE_ASYNC_LDS
    if (tid < 32) {  // wave 0: 2 x b128 per lane covers 1 KB
        const float* g0 = pred + pbase;
        async_copy16(g0 + lane * 4, &rowbuf[0][lane * 4]);
        async_copy16(g0 + 128 + lane * 4, &rowbuf[0][128 + lane * 4]);
    }
    wait_async0();  // row 0 landed (only wave 0 has a nonzero ASYNCcnt)
    if (tid < 32 && Tin > 1) {
        const float* g1 = pred + pbase + (size_t)C;
        async_copy16(g1 + lane * 4, &rowbuf[1][lane * 4]);
        async_copy16(g1 + 128 + lane * 4, &rowbuf[1][128 + lane * 4]);
    }
#else
    if (tid < C) rowbuf[0][tid] = pred[pbase + tid];
    float pre = 0.0f;
    if (tid < C && Tin > 1) pre = pred[pbase + (size_t)C + tid];
#endif
    __syncthreads();  // row 0 + skipS visible to everyone

    const float n0 = norm[(size_t)b * T];
    for (int s = tid; s < S; s += blockDim.x) {
        float a = NEGV;
        if (s == 0) a = rowbuf[0][blank] - n0;
        else if (s == 1) a = rowbuf[0][extS[1]] - n0;
        abuf[0][s] = a;
    }

    int cur = 0;
    for (int t = 1; t < Tin; ++t) {
        const int rb = t & 1;
#if HAVE_ASYNC_LDS
        wait_async0();    // row t's async copies complete (wave 0)
        __syncthreads();  // publish row t + abuf[cur]; all readers of
                          // rowbuf[rb^1] from step t-1 are done
        if (tid < 32 && (t + 1) < Tin) {  // prefetch row t+1 into other buf
            const float* gn = pred + pbase + (size_t)(t + 1) * C;
            float* ln = &rowbuf[rb ^ 1][0];
            async_copy16(gn + lane * 4, ln + lane * 4);
            async_copy16(gn + 128 + lane * 4, ln + 128 + lane * 4);
        }
#else
        if (tid < C) rowbuf[rb][tid] = pre;  // publish row t
        __syncthreads();
        if (tid < C && (t + 1) < Tin)
            pre = pred[pbase + (size_t)(t + 1) * C + tid];  // overlap latency
#endif
        const float nt = norm[(size_t)b * T + t];
        const int nxt = cur ^ 1;
        if (tid < S) {
            const int s = tid;
            float a1 = abuf[cur][s];
            float a2 = (s >= 1) ? abuf[cur][s - 1] : NEGV;
            float a3 = (s >= 2 && skipS[s]) ? abuf[cur][s - 2] : NEGV;
            float e = rowbuf[rb][extS[s]] - nt;
            abuf[nxt][s] = logaddexp3(a1, a2, a3) + e;
        }
        cur = nxt;
    }
    __syncthreads();

    if (tid == 0) {
        int ll = lab_len[b];
        if (ll < 1) ll = 1;
        if (ll > L) ll = L;
        const float f0 = abuf[cur][2 * ll - 1];
        const float f1 = abuf[cur][2 * ll];
        const float m = fmaxf(f0, f1);
        loss[b] = -(m + __logf(__expf(f0 - m) + __expf(f1 - m)));  // scale 1.0
    }
}

// ---------------------------------------------------------------------------
extern "C" void kernel_launch(void* const* d_in, const int* in_sizes, int n_in,
                              void* d_out, int out_size, void* d_ws,
                              size_t ws_size, hipStream_t stream) {
    const float* pred   = (const float*)d_in[0];
    const int*   in_len = (const int*)d_in[1];
    const int*   labels = (const int*)d_in[2];
    const int*   lablen = (const int*)d_in[3];
    float*       out    = (float*)d_out;
    float*       norm   = (float*)d_ws;  // B*T floats (512 KB for this shape)

    constexpr int C = 256;
    const int B = in_sizes[1];                 // input_lengths is [B,1]
    const int L = in_sizes[2] / B;             // labels is [B,L]
    const int T = in_sizes[0] / (B * C);       // predictions is [B,T,C]
    const int nrows = B * T;

    lse_norm_kernel<C><<<(nrows + 63) / 64, 128, 0, stream>>>(pred, norm, nrows);
    ctc_scan_kernel<<<B, 544, 0, stream>>>(pred, norm, labels, in_len, lablen,
                                           out, T, C, L);
}